// DynamicGraph_4836133175698
// MI455X (gfx1250) — compile-verified
//
#include <hip/hip_runtime.h>
#include <math.h>

typedef __attribute__((ext_vector_type(2))) float v2f;
typedef __attribute__((ext_vector_type(8))) float v8f;

#define STATE_DIM 256
#define PROJ_DIM  64
#define NCOL      4096
#define NROWS     8192          // B*N = 2*4096
#define TOPK      8
#define LDS_STRIDE 4100         // 4096 + 4 pad: conflict-free C writes, 16B-aligned rows

// ---------------------------------------------------------------------------
// Kernel A: Q = X*Wq^T + bq ; K = X*Wk^T + bk   via V_WMMA_F32_16X16X4_F32
// One wave per 16x16 output tile. 8 waves / block.
// Tiles: 2 (Q,K) * 512 rowTiles * 4 colTiles = 4096 tiles -> 512 blocks.
// ---------------------------------------------------------------------------
__global__ __launch_bounds__(256) void proj_kernel(
    const float* __restrict__ X,   // (8192,256)
    const float* __restrict__ Wq,  // (64,256)
    const float* __restrict__ bq,  // (64)
    const float* __restrict__ Wk,  // (64,256)
    const float* __restrict__ bk,  // (64)
    float* __restrict__ Qout,      // (8192,64)
    float* __restrict__ Kout)      // (8192,64)
{
  const int lane = threadIdx.x & 31;
  const int wid  = threadIdx.x >> 5;
  const int tile = blockIdx.x * 8 + wid;

  const int which = tile >> 11;           // 0 = Q, 1 = K
  const int rt    = (tile & 2047) >> 2;   // 0..511
  const int ct    = tile & 3;             // 0..3

  const float* __restrict__ W    = which ? Wk : Wq;
  const float* __restrict__ bias = which ? bk : bq;
  float* __restrict__ Out        = which ? Kout : Qout;

  const int halfM = lane & 15;
  const int koff  = (lane >> 4) * 2;      // half-wave holds K pair {0,1} or {2,3}

  const float* xrow = X + (size_t)(rt * 16 + halfM) * STATE_DIM + koff;
  const float* wrow = W + (size_t)(ct * 16 + halfM) * STATE_DIM + koff;

  v8f acc = {};
#pragma unroll
  for (int kb = 0; kb < STATE_DIM; kb += 4) {
    v2f a = *(const v2f*)(xrow + kb);   // A: 16x4 of X
    v2f b = *(const v2f*)(wrow + kb);   // B: 4x16 of W^T (B[k][n] = W[n][k])
    acc = __builtin_amdgcn_wmma_f32_16x16x4_f32(
        false, a, false, b, (short)0, acc, false, false);
  }

  const int   nOut  = ct * 16 + halfM;
  const float bv    = bias[nOut];
  const int   mBase = rt * 16 + ((lane >> 4) << 3);   // lanes 16-31 hold M=8..15
#pragma unroll
  for (int r = 0; r < 8; ++r) {
    Out[(size_t)(mBase + r) * PROJ_DIM + nOut] = acc[r] + bv;
  }
}

// ---------------------------------------------------------------------------
// Kernel B: per 16-row block: scores (16x4096) into 256KB LDS via WMMA,
// then per-row top-8 + softmax, single coalesced write of the output rows.
// 512 blocks x 512 threads (16 waves); dynamic LDS = 16*4100*4 = 262400 B.
// ---------------------------------------------------------------------------
__global__ __launch_bounds__(512) void score_kernel(
    const float* __restrict__ Q,        // (8192,64)
    const float* __restrict__ K,        // (8192,64)
    const float* __restrict__ log_temp, // scalar
    float* __restrict__ out)            // (2,4096,4096)
{
  extern __shared__ float S[];          // [16][LDS_STRIDE]

  const int lane  = threadIdx.x & 31;
  const int wid   = threadIdx.x >> 5;   // 0..15
  const int halfM = lane & 15;
  const int koff  = (lane >> 4) * 2;

  const int rowBlock = blockIdx.x;             // 0..511
  const int batch    = rowBlock >> 8;          // 256 row-blocks per batch

  // --- Phase 1: S = Q_tile (16x64) * K_b^T (64x4096), WMMA f32 16x16x4 ---
  v2f afrag[16];
  {
    const float* qrow = Q + (size_t)(rowBlock * 16 + halfM) * PROJ_DIM + koff;
#pragma unroll
    for (int i = 0; i < 16; ++i) afrag[i] = *(const v2f*)(qrow + i * 4);
  }
  const float* Kb = K + (size_t)batch * NCOL * PROJ_DIM;

  for (int ctile = wid; ctile < NCOL / 16; ctile += 16) {
    const float* krow = Kb + (size_t)(ctile * 16 + halfM) * PROJ_DIM + koff;
    v8f acc = {};
#pragma unroll
    for (int i = 0; i < 16; ++i) {
      v2f b = *(const v2f*)(krow + i * 4);   // B[k][n] = K[n][k]
      acc = __builtin_amdgcn_wmma_f32_16x16x4_f32(
          false, afrag[i], false, b, (short)0, acc, false, false);
    }
    const int n     = ctile * 16 + halfM;
    const int mBase = (lane >> 4) << 3;
#pragma unroll
    for (int r = 0; r < 8; ++r) S[(mBase + r) * LDS_STRIDE + n] = acc[r];
  }
  __syncthreads();

  // --- Phase 2: top-8 + softmax; one row per wave ---
  float t = expf(log_temp[0]);
  t = fminf(fmaxf(t, 0.1f), 10.0f);
  const float invscale = 1.0f / (8.0f * t);    // 1 / (sqrt(64) * temperature)

  const int m = wid;                            // row 0..15
  float* srow = &S[m * LDS_STRIDE];

  float topv[TOPK];
  int   topi[TOPK];
#pragma unroll
  for (int kth = 0; kth < TOPK; ++kth) {
    float best = -INFINITY;
    int   bidx = 0x7fffffff;
    // float4 scan: row base is 16B-aligned (LDS_STRIDE*4 % 16 == 0)
    for (int c4 = lane * 4; c4 < NCOL; c4 += 32 * 4) {
      float4 v = *(const float4*)(srow + c4);
      if (v.x > best) { best = v.x; bidx = c4; }
      if (v.y > best) { best = v.y; bidx = c4 + 1; }
      if (v.z > best) { best = v.z; bidx = c4 + 2; }
      if (v.w > best) { best = v.w; bidx = c4 + 3; }
    }
    // butterfly reduce; ties -> lower index (matches jax.lax.top_k)
#pragma unroll
    for (int off = 16; off > 0; off >>= 1) {
      float ov = __shfl_xor(best, off, 32);
      int   oi = __shfl_xor(bidx, off, 32);
      if (ov > best || (ov == best && oi < bidx)) { best = ov; bidx = oi; }
    }
    topv[kth] = best;
    topi[kth] = bidx;
    if (lane == 0) srow[bidx] = -INFINITY;   // mask for next pass
  }

  // softmax over the 8 kept scores (everything else underflows to 0.0,
  // exactly matching exp(-1e9 - max) == 0.0f in the reference)
  const float mx = topv[0] * invscale;       // first pick is the max
  float p[TOPK];
  float sum = 0.0f;
#pragma unroll
  for (int j = 0; j < TOPK; ++j) {
    p[j] = expf(topv[j] * invscale - mx);
    sum += p[j];
  }
  const float inv = 1.0f / sum;
#pragma unroll
  for (int j = 0; j < TOPK; ++j) p[j] *= inv;

  // single coalesced pass over the output row: zeros + scattered probs
  float* orow = out + (size_t)(rowBlock * 16 + m) * NCOL;
  for (int c4 = lane * 4; c4 < NCOL; c4 += 32 * 4) {
    float4 v = make_float4(0.f, 0.f, 0.f, 0.f);
#pragma unroll
    for (int j = 0; j < TOPK; ++j) {
      const int idx = topi[j];
      const float pv = p[j];
      if (idx == c4)     v.x = pv;
      if (idx == c4 + 1) v.y = pv;
      if (idx == c4 + 2) v.z = pv;
      if (idx == c4 + 3) v.w = pv;
    }
    *(float4*)(orow + c4) = v;
  }
}

// ---------------------------------------------------------------------------
extern "C" void kernel_launch(void* const* d_in, const int* in_sizes, int n_in,
                              void* d_out, int out_size, void* d_ws, size_t ws_size,
                              hipStream_t stream) {
  const float* X  = (const float*)d_in[0];  // action_states (2,4096,256)
  const float* Wq = (const float*)d_in[1];
  const float* bq = (const float*)d_in[2];
  const float* Wk = (const float*)d_in[3];
  const float* bk = (const float*)d_in[4];
  const float* lt = (const float*)d_in[5];  // log_temperature scalar

  float* Qws = (float*)d_ws;                       // 8192*64 floats = 2 MB
  float* Kws = Qws + (size_t)NROWS * PROJ_DIM;     // next 2 MB

  const int ldsBytes = 16 * LDS_STRIDE * 4;        // 262400 B (< 320KB/WGP)
  (void)hipFuncSetAttribute((const void*)score_kernel,
                            hipFuncAttributeMaxDynamicSharedMemorySize,
                            ldsBytes);

  proj_kernel<<<512, 256, 0, stream>>>(X, Wq, bq, Wk, bk, Qws, Kws);
  score_kernel<<<512, 512, ldsBytes, stream>>>(Qws, Kws, lt, (float*)d_out);
}